// VanillaGCN_25615184953981
// MI455X (gfx1250) — compile-verified
//
#include <hip/hip_runtime.h>
#include <math.h>

// Problem constants (match reference)
#define BB 32
#define TT 1024
#define DD 256
#define LN_EPS 1e-5f
#define EPS_ADJ 1e-6f

typedef __attribute__((ext_vector_type(2))) float v2f;
typedef __attribute__((ext_vector_type(8))) float v8f;

static __device__ __forceinline__ float wave_reduce_sum(float v) {
#pragma unroll
  for (int m = 16; m >= 1; m >>= 1) v += __shfl_xor(v, m, 32);
  return v;
}

// ---------------------------------------------------------------------------
// 1) Row L2-normalize: xn[b,t,:] = x / max(||x||, 1e-12)
//    block = (32,8): one wave per row
// ---------------------------------------------------------------------------
__global__ void gcn_rownorm(const float* __restrict__ x, float* __restrict__ xn) {
  const int lane = threadIdx.x;
  const size_t row = (size_t)blockIdx.x * 8 + threadIdx.y;  // B*T rows
  const float* xr = x + row * DD;
  float vals[8];
  float s = 0.f;
#pragma unroll
  for (int u = 0; u < 8; ++u) {
    vals[u] = xr[u * 32 + lane];
    s += vals[u] * vals[u];
  }
  s = wave_reduce_sum(s);
  const float inv = 1.0f / fmaxf(sqrtf(s), 1e-12f);
  float* xo = xn + row * DD;
#pragma unroll
  for (int u = 0; u < 8; ++u) xo[u * 32 + lane] = vals[u] * inv;
}

// ---------------------------------------------------------------------------
// 2) sim[b] = xn[b] @ xn[b]^T  (masked), f32 WMMA 16x16x4.
//    Each wave computes a 16x64 strip (4 N-tiles) so the A fragment is
//    loaded once per K-step and feeds 4 WMMAs.
//    grid = (T/256, T/16, B), block = (32,4)
// ---------------------------------------------------------------------------
__global__ void gcn_sim_wmma(const float* __restrict__ xn,
                             const int* __restrict__ mask,
                             float* __restrict__ sim) {
  const int b = blockIdx.z;
  const int i0 = blockIdx.y * 16;
  const int jbase = (blockIdx.x * 16 + threadIdx.y * 4) * 16;  // 4 tiles
  const int lane = threadIdx.x;
  const int half = lane >> 4;
  const int l16 = lane & 15;

  const float* Xb = xn + (size_t)b * TT * DD;
  const float* arow = Xb + (size_t)(i0 + l16) * DD + half * 2;
  const float* brow[4];
#pragma unroll
  for (int jt = 0; jt < 4; ++jt)
    brow[jt] = Xb + (size_t)(jbase + jt * 16 + l16) * DD + half * 2;

  v8f c[4] = {};
#pragma unroll 4
  for (int k = 0; k < DD; k += 4) {
    const v2f a = *(const v2f*)(arow + k);
#pragma unroll
    for (int jt = 0; jt < 4; ++jt) {
      const v2f bv = *(const v2f*)(brow[jt] + k);
      c[jt] = __builtin_amdgcn_wmma_f32_16x16x4_f32(false, a, false, bv,
                                                    (short)0, c[jt], false,
                                                    false);
    }
  }

  const int* mb = mask + (size_t)b * TT;
  float* S = sim + ((size_t)b * TT + i0) * TT;
#pragma unroll
  for (int jt = 0; jt < 4; ++jt) {
    const float mj = (float)mb[jbase + jt * 16 + l16];
#pragma unroll
    for (int r = 0; r < 8; ++r) {
      const int mrow = r + half * 8;
      const float mi = (float)mb[i0 + mrow];
      S[(size_t)mrow * TT + jbase + jt * 16 + l16] = c[jt][r] * mi * mj;
    }
  }
}

// ---------------------------------------------------------------------------
// 3) Per-row top-2 of sim with diagonal suppressed to -1.0.
//    Tie-break: lower index wins (matches jax.lax.top_k).
//    block = (32,8): one wave per row.
// ---------------------------------------------------------------------------
__global__ void gcn_top2(const float* __restrict__ sim, int* __restrict__ topk) {
  const int lane = threadIdx.x;
  const size_t row = (size_t)blockIdx.x * 8 + threadIdx.y;  // b*T + i
  const int i = (int)(row & (TT - 1));
  const float* srow = sim + row * TT;

  float v1 = -1e30f, v2 = -1e30f;
  int i1 = 0x7fffffff, i2 = 0x7fffffff;
  for (int j = lane; j < TT; j += 32) {
    const float v = (j == i) ? -1.0f : srow[j];
    if (v > v1 || (v == v1 && j < i1)) {
      v2 = v1; i2 = i1; v1 = v; i1 = j;
    } else if (v > v2 || (v == v2 && j < i2)) {
      v2 = v; i2 = j;
    }
  }
#pragma unroll
  for (int m = 16; m >= 1; m >>= 1) {
    const float ov1 = __shfl_xor(v1, m, 32);
    const int   oi1 = __shfl_xor(i1, m, 32);
    const float ov2 = __shfl_xor(v2, m, 32);
    const int   oi2 = __shfl_xor(i2, m, 32);
    float nv1, nv2; int ni1, ni2;
    const bool firstOurs = (v1 > ov1) || (v1 == ov1 && i1 < oi1);
    if (firstOurs) {
      nv1 = v1; ni1 = i1;
      const bool secOurs = (v2 > ov1) || (v2 == ov1 && i2 < oi1);
      nv2 = secOurs ? v2 : ov1; ni2 = secOurs ? i2 : oi1;
    } else {
      nv1 = ov1; ni1 = oi1;
      const bool secTheirs = (ov2 > v1) || (ov2 == v1 && oi2 < i1);
      nv2 = secTheirs ? ov2 : v1; ni2 = secTheirs ? oi2 : i1;
    }
    v1 = nv1; i1 = ni1; v2 = nv2; i2 = ni2;
  }
  if (lane == 0) {
    topk[row * 2 + 0] = i1;
    topk[row * 2 + 1] = i2;
  }
}

// ---------------------------------------------------------------------------
// 4) Build adjacency in place over sim; pre-norm adj is symmetric, so
//    col_sum == row_sum -> only row sums needed. block = (32,8).
// ---------------------------------------------------------------------------
__global__ void gcn_adj_build(float* __restrict__ sim,
                              const int* __restrict__ topk,
                              const int* __restrict__ mask,
                              float* __restrict__ row_sum) {
  const int lane = threadIdx.x;
  const size_t row = (size_t)blockIdx.x * 8 + threadIdx.y;
  const int b = (int)(row >> 10);
  const int i = (int)(row & (TT - 1));
  float* srow = sim + row * TT;
  const int* tk = topk + (size_t)b * TT * 2;
  const int* mb = mask + (size_t)b * TT;
  const int ti0 = tk[2 * i], ti1 = tk[2 * i + 1];
  const float mi = (float)mb[i];

  float acc = 0.f;
  for (int j = lane; j < TT; j += 32) {
    const float sv = srow[j];
    const int d = j - i;
    const float base = (d >= -1 && d <= 1) ? 1.f : 0.f;
    const int tj0 = tk[2 * j], tj1 = tk[2 * j + 1];
    float ext = ((ti0 == j) | (ti1 == j) | (tj0 == i) | (tj1 == i)) ? 1.f : 0.f;
    ext *= mi * (float)mb[j];
    const float e = fminf(base + ext, 1.f);
    const float a = e * sv + ((j == i) ? 1.f : 0.f);
    srow[j] = a;
    acc += a;
  }
  acc = wave_reduce_sum(acc);
  if (lane == 0) row_sum[row] = acc;
}

// ---------------------------------------------------------------------------
// 5) Symmetric normalization: adj /= sqrt((rs_i+eps)(rs_j+eps)), times mask.
// ---------------------------------------------------------------------------
__global__ void gcn_adj_norm(float* __restrict__ adj,
                             const float* __restrict__ row_sum,
                             const int* __restrict__ mask) {
  const size_t total = (size_t)BB * TT * TT;
  for (size_t idx = (size_t)blockIdx.x * blockDim.x + threadIdx.x; idx < total;
       idx += (size_t)gridDim.x * blockDim.x) {
    const size_t j = idx & (TT - 1);
    const size_t bi = idx >> 10;       // b*T + i
    const size_t b = bi >> 10;
    const size_t i = bi & (TT - 1);
    const float rs_i = row_sum[bi] + EPS_ADJ;
    const float rs_j = row_sum[b * TT + j] + EPS_ADJ;
    const float mi = (float)mask[b * TT + i];
    const float mj = (float)mask[b * TT + j];
    adj[idx] = adj[idx] * rsqrtf(rs_i * rs_j) * mi * mj;
  }
}

// ---------------------------------------------------------------------------
// 6) hout[b] = adj[b] @ hin[b]   (T x T) x (T x D), f32 WMMA.
//    Each wave computes a 16x64 strip (4 N-tiles); one (32,4) block covers
//    all 16 N tiles (N = D = 256), so the adj row strip (the 128 MB operand)
//    is loaded once per K-step per wave and feeds 4 WMMAs.
//    grid = (1, T/16, B), block = (32,4)
// ---------------------------------------------------------------------------
__global__ void gcn_spmm_wmma(const float* __restrict__ adj,
                              const float* __restrict__ hin,
                              float* __restrict__ hout) {
  const int b = blockIdx.z;
  const int i0 = blockIdx.y * 16;
  const int nbase = threadIdx.y * 64;  // 4 tiles of 16
  const int lane = threadIdx.x;
  const int half = lane >> 4;
  const int l16 = lane & 15;

  const float* A = adj + (size_t)b * TT * TT;
  const float* Bm = hin + (size_t)b * TT * DD;
  const float* arow = A + (size_t)(i0 + l16) * TT + half * 2;
  const float* bcol[4];
#pragma unroll
  for (int jt = 0; jt < 4; ++jt)
    bcol[jt] = Bm + (size_t)(half * 2) * DD + nbase + jt * 16 + l16;

  v8f c[4] = {};
#pragma unroll 2
  for (int k = 0; k < TT; k += 4) {
    const v2f a = *(const v2f*)(arow + k);
#pragma unroll
    for (int jt = 0; jt < 4; ++jt) {
      v2f bv;
      bv.x = bcol[jt][(size_t)k * DD];
      bv.y = bcol[jt][(size_t)(k + 1) * DD];
      c[jt] = __builtin_amdgcn_wmma_f32_16x16x4_f32(false, a, false, bv,
                                                    (short)0, c[jt], false,
                                                    false);
    }
  }
  float* O = hout + (size_t)b * TT * DD;
#pragma unroll
  for (int jt = 0; jt < 4; ++jt)
#pragma unroll
    for (int r = 0; r < 8; ++r)
      O[(size_t)(i0 + r + half * 8) * DD + nbase + jt * 16 + l16] = c[jt][r];
}

// ---------------------------------------------------------------------------
// 7) y = hin @ W^T   rows = B*T, W is [out, in] torch layout (row-major), so
//    B-fragment reads W[n, k] contiguously. Each wave computes 4 N-tiles.
//    grid = (1, B*T/16), block = (32,4)
// ---------------------------------------------------------------------------
__global__ void gcn_dense_wmma(const float* __restrict__ hin,
                               const float* __restrict__ W,
                               float* __restrict__ y) {
  const int i0 = blockIdx.y * 16;      // over B*T rows
  const int nbase = threadIdx.y * 64;  // 4 tiles of 16
  const int lane = threadIdx.x;
  const int half = lane >> 4;
  const int l16 = lane & 15;

  const float* arow = hin + (size_t)(i0 + l16) * DD + half * 2;
  const float* brow[4];
#pragma unroll
  for (int jt = 0; jt < 4; ++jt)
    brow[jt] = W + (size_t)(nbase + jt * 16 + l16) * DD + half * 2;

  v8f c[4] = {};
#pragma unroll 4
  for (int k = 0; k < DD; k += 4) {
    const v2f a = *(const v2f*)(arow + k);
#pragma unroll
    for (int jt = 0; jt < 4; ++jt) {
      const v2f bv = *(const v2f*)(brow[jt] + k);
      c[jt] = __builtin_amdgcn_wmma_f32_16x16x4_f32(false, a, false, bv,
                                                    (short)0, c[jt], false,
                                                    false);
    }
  }
#pragma unroll
  for (int jt = 0; jt < 4; ++jt)
#pragma unroll
    for (int r = 0; r < 8; ++r)
      y[(size_t)(i0 + r + half * 8) * DD + nbase + jt * 16 + l16] = c[jt][r];
}

// ---------------------------------------------------------------------------
// 8) Fused bias + ELU + LayerNorm per row. block = (32,8): one wave per row.
// ---------------------------------------------------------------------------
__global__ void gcn_epilogue(const float* __restrict__ y,
                             const float* __restrict__ bias,
                             const float* __restrict__ g,
                             const float* __restrict__ be,
                             float* __restrict__ out) {
  const int lane = threadIdx.x;
  const size_t row = (size_t)blockIdx.x * 8 + threadIdx.y;
  const float* yr = y + row * DD;
  float v[8];
  float s = 0.f;
#pragma unroll
  for (int u = 0; u < 8; ++u) {
    const int idx = u * 32 + lane;
    float t = yr[idx] + bias[idx];
    t = (t > 0.f) ? t : (expf(t) - 1.f);  // ELU alpha=1
    v[u] = t;
    s += t;
  }
  const float mu = wave_reduce_sum(s) * (1.0f / DD);
  float var = 0.f;
#pragma unroll
  for (int u = 0; u < 8; ++u) {
    const float d = v[u] - mu;
    var += d * d;
  }
  var = wave_reduce_sum(var) * (1.0f / DD);
  const float inv = rsqrtf(var + LN_EPS);
  float* orow = out + row * DD;
#pragma unroll
  for (int u = 0; u < 8; ++u) {
    const int idx = u * 32 + lane;
    orow[idx] = (v[u] - mu) * inv * g[idx] + be[idx];
  }
}

// ---------------------------------------------------------------------------
// Host-side launch
// ---------------------------------------------------------------------------
extern "C" void kernel_launch(void* const* d_in, const int* in_sizes, int n_in,
                              void* d_out, int out_size, void* d_ws,
                              size_t ws_size, hipStream_t stream) {
  const float* x = (const float*)d_in[0];
  const int* mask = (const int*)d_in[1];
  const float* W0 = (const float*)d_in[2];
  const float* b0 = (const float*)d_in[3];
  const float* g0 = (const float*)d_in[4];
  const float* be0 = (const float*)d_in[5];
  const float* W1 = (const float*)d_in[6];
  const float* b1 = (const float*)d_in[7];
  const float* g1 = (const float*)d_in[8];
  const float* be1 = (const float*)d_in[9];
  float* out = (float*)d_out;

  // Workspace layout (floats)
  const size_t n_xtd = (size_t)BB * TT * DD;  // 8M floats
  const size_t n_sim = (size_t)BB * TT * TT;  // 32M floats
  float* ws = (float*)d_ws;
  float* xn = ws;                      // 8M
  float* sim = xn + n_xtd;             // 32M (becomes adj in place)
  float* htmp = sim + n_sim;           // 8M
  float* ybuf = htmp + n_xtd;          // 8M
  float* hbuf = ybuf + n_xtd;          // 8M
  float* row_sum = hbuf + n_xtd;       // 32768
  int* topk = (int*)(row_sum + (size_t)BB * TT);  // 65536 ints

  const int rows = BB * TT;  // 32768

  // 1) row normalize
  gcn_rownorm<<<rows / 8, dim3(32, 8), 0, stream>>>(x, xn);

  // 2) similarity GEMM (WMMA f32), 16x64 strip per wave
  gcn_sim_wmma<<<dim3(TT / 256, TT / 16, BB), dim3(32, 4), 0, stream>>>(
      xn, mask, sim);

  // 3) top-2 per row
  gcn_top2<<<rows / 8, dim3(32, 8), 0, stream>>>(sim, topk);

  // 4) adjacency build in place + row sums
  gcn_adj_build<<<rows / 8, dim3(32, 8), 0, stream>>>(sim, topk, mask, row_sum);

  // 5) symmetric normalization
  gcn_adj_norm<<<8192, 256, 0, stream>>>(sim, row_sum, mask);

  // --- layer 0 ---
  gcn_spmm_wmma<<<dim3(1, TT / 16, BB), dim3(32, 4), 0, stream>>>(sim, x, htmp);
  gcn_dense_wmma<<<dim3(1, rows / 16), dim3(32, 4), 0, stream>>>(htmp, W0, ybuf);
  gcn_epilogue<<<rows / 8, dim3(32, 8), 0, stream>>>(ybuf, b0, g0, be0, hbuf);

  // --- layer 1 ---
  gcn_spmm_wmma<<<dim3(1, TT / 16, BB), dim3(32, 4), 0, stream>>>(sim, hbuf,
                                                                  htmp);
  gcn_dense_wmma<<<dim3(1, rows / 16), dim3(32, 4), 0, stream>>>(htmp, W1, ybuf);
  gcn_epilogue<<<rows / 8, dim3(32, 8), 0, stream>>>(ybuf, b1, g1, be1, out);
}